// HashEncoder_38216619000480
// MI455X (gfx1250) — compile-verified
//
#include <hip/hip_runtime.h>
#include <stdint.h>

// Instant-NGP style multi-resolution hash-grid encoder for MI455X (gfx1250).
//
//  * 16 tables x 4MB = 64MB -> resident in 192MB L2; 1.07GB of hashed gathers
//    are L2 hits. HBM moves ~210MB (~9us floor @ 23.3 TB/s).
//  * Output (132MB) streamed with TH_STORE_NT so it never evicts the tables.
//  * xyz staged block-wide into LDS via the gfx1250 async global->LDS path
//    (ASYNCcnt + s_wait_asynccnt), then broadcast-read (conflict-free same-
//    address LDS reads) by the 16 level-lanes of each point.
//  * WMMA is inapplicable: per-lane divergent gathers, no shared matmul operand.
//  * wave32 mapping: lanes 0..15 = levels 0..15 of point P, lanes 16..31 =
//    point P+1 -> each wave's output store is one contiguous 256B burst.

typedef float v2f __attribute__((ext_vector_type(2)));
typedef __attribute__((address_space(1))) int gint_t;   // global int
typedef __attribute__((address_space(3))) int lint_t;   // LDS int

#define N_LEVELS   16
#define LOG2_T     19
#define TABLE_SIZE (1u << LOG2_T)
#define HASH_MASK  (TABLE_SIZE - 1u)
#define PRIME_Y    2654435761u
#define PRIME_Z    805459861u
#define PTS_PER_BLOCK 16u   // 256 threads / 16 levels

#if defined(__has_builtin)
#if __has_builtin(__builtin_amdgcn_global_load_async_to_lds_b32)
#define HAVE_ASYNC_LDS 1
#endif
#if __has_builtin(__builtin_amdgcn_s_wait_asynccnt)
#define HAVE_WAIT_ASYNC 1
#endif
#endif

// {grid_size, 1/grid_size} packed so one global_load_b64 fetches both.
// res_i = floor(16 * 2^(i/3)), i = 0..15  (base 16 -> finest 512)
__device__ __constant__ v2f kGsInv[N_LEVELS] = {
    {2.0f/16.0f,  8.0f},   {2.0f/20.0f,  10.0f},
    {2.0f/25.0f,  12.5f},  {2.0f/32.0f,  16.0f},
    {2.0f/40.0f,  20.0f},  {2.0f/50.0f,  25.0f},
    {2.0f/64.0f,  32.0f},  {2.0f/80.0f,  40.0f},
    {2.0f/101.0f, 50.5f},  {2.0f/128.0f, 64.0f},
    {2.0f/161.0f, 80.5f},  {2.0f/203.0f, 101.5f},
    {2.0f/256.0f, 128.0f}, {2.0f/322.0f, 161.0f},
    {2.0f/406.0f, 203.0f}, {2.0f/512.0f, 256.0f} };

__device__ __forceinline__ v2f lerp2(v2f a, v2f b, float w) {
    return a + w * (b - a);   // fma form of a*(1-w) + b*w
}

__global__ __launch_bounds__(256)
void hash_encode_kernel(const float* __restrict__ xyz,
                        const float* __restrict__ tables,
                        float* __restrict__ out,
                        float* __restrict__ mask,
                        uint32_t n_points)
{
    __shared__ float s_xyz[PTS_PER_BLOCK * 3];

    const uint32_t t          = threadIdx.x;
    const uint32_t base_point = blockIdx.x * PTS_PER_BLOCK;
    const uint32_t n_dw_total = 3u * n_points;

    // ---- Stage this block's 16 points (48 dwords) into LDS ----------------
    const uint32_t gidx = base_point * 3u + t;
    if (t < PTS_PER_BLOCK * 3u && gidx < n_dw_total) {
#if defined(HAVE_ASYNC_LDS)
        __builtin_amdgcn_global_load_async_to_lds_b32(
            (gint_t*)(xyz + gidx),
            (lint_t*)(&s_xyz[t]),
            /*offset=*/0, /*cpol=*/0);
#else
        s_xyz[t] = __builtin_nontemporal_load(xyz + gidx);
#endif
    }
#if defined(HAVE_ASYNC_LDS)
#if defined(HAVE_WAIT_ASYNC)
    __builtin_amdgcn_s_wait_asynccnt(0);
#else
    asm volatile("s_wait_asynccnt 0x0" ::: "memory");
#endif
#endif
    __syncthreads();

    const uint32_t level = t & (N_LEVELS - 1u);
    const uint32_t lp    = t >> 4;               // local point 0..15
    const uint32_t point = base_point + lp;
    if (point >= n_points) return;

    // Same-address LDS reads for the 16 lanes of one point -> broadcast.
    const float x = s_xyz[lp * 3u + 0u];
    const float y = s_xyz[lp * 3u + 1u];
    const float z = s_xyz[lp * 3u + 2u];

    const float xc = fminf(fmaxf(x, -1.0f), 1.0f);
    const float yc = fminf(fmaxf(y, -1.0f), 1.0f);
    const float zc = fminf(fmaxf(z, -1.0f), 1.0f);

    const v2f   gi  = kGsInv[level];
    const float gs  = gi.x;
    const float inv = gi.y;

    // bl = floor((p - BOX_MIN)/gs); vmin = bl*gs + BOX_MIN; w = (p - vmin)/gs
    const float blx = floorf((xc + 1.0f) * inv);
    const float bly = floorf((yc + 1.0f) * inv);
    const float blz = floorf((zc + 1.0f) * inv);

    const float wx = (xc - (blx * gs - 1.0f)) * inv;
    const float wy = (yc - (bly * gs - 1.0f)) * inv;
    const float wz = (zc - (blz * gs - 1.0f)) * inv;

    const uint32_t ix = (uint32_t)blx;
    const uint32_t iy = (uint32_t)bly;
    const uint32_t iz = (uint32_t)blz;

    // Partial hash products per axis (primes {1, 2654435761, 805459861}).
    const uint32_t hx0 = ix;
    const uint32_t hx1 = ix + 1u;
    const uint32_t hy0 = iy * PRIME_Y;
    const uint32_t hy1 = (iy + 1u) * PRIME_Y;
    const uint32_t hz0 = iz * PRIME_Z;
    const uint32_t hz1 = (iz + 1u) * PRIME_Z;

    const v2f* __restrict__ tbl =
        (const v2f*)tables + (size_t)level * TABLE_SIZE;

    // Corner order matches reference OFFSETS: index = 4*i + 2*j + k.
    // All 8 gathers issued back-to-back -> one s_wait_loadcnt batch.
    const v2f e000 = tbl[(hx0 ^ hy0 ^ hz0) & HASH_MASK];
    const v2f e001 = tbl[(hx0 ^ hy0 ^ hz1) & HASH_MASK];
    const v2f e010 = tbl[(hx0 ^ hy1 ^ hz0) & HASH_MASK];
    const v2f e011 = tbl[(hx0 ^ hy1 ^ hz1) & HASH_MASK];
    const v2f e100 = tbl[(hx1 ^ hy0 ^ hz0) & HASH_MASK];
    const v2f e101 = tbl[(hx1 ^ hy0 ^ hz1) & HASH_MASK];
    const v2f e110 = tbl[(hx1 ^ hy1 ^ hz0) & HASH_MASK];
    const v2f e111 = tbl[(hx1 ^ hy1 ^ hz1) & HASH_MASK];

    const v2f c00 = lerp2(e000, e100, wx);
    const v2f c01 = lerp2(e001, e101, wx);
    const v2f c10 = lerp2(e010, e110, wx);
    const v2f c11 = lerp2(e011, e111, wx);
    const v2f c0  = lerp2(c00, c10, wy);
    const v2f c1  = lerp2(c01, c11, wy);
    const v2f r   = lerp2(c0, c1, wz);

    // Contiguous 256B per wave; non-temporal (output is write-once stream).
    __builtin_nontemporal_store(
        r, (v2f*)(out + (size_t)point * (N_LEVELS * 2u) + level * 2u));

    if (level == 0u) {
        const float keep =
            (x == xc && y == yc && z == zc) ? 1.0f : 0.0f;
        __builtin_nontemporal_store(keep, mask + point);
    }
}

extern "C" void kernel_launch(void* const* d_in, const int* in_sizes, int n_in,
                              void* d_out, int out_size, void* d_ws, size_t ws_size,
                              hipStream_t stream)
{
    const float* xyz    = (const float*)d_in[0];   // (N, 3) f32
    const float* tables = (const float*)d_in[1];   // (16, 2^19, 2) f32
    float*       out    = (float*)d_out;           // N*32 features, then N mask

    const uint32_t n_points = (uint32_t)(in_sizes[0] / 3);
    float* mask = out + (size_t)n_points * (N_LEVELS * 2u);

    const uint32_t blocks = (n_points + PTS_PER_BLOCK - 1u) / PTS_PER_BLOCK;

    hipLaunchKernelGGL(hash_encode_kernel, dim3(blocks), dim3(256), 0, stream,
                       xyz, tables, out, mask, n_points);
}